// KNN_35639638622492
// MI455X (gfx1250) — compile-verified
//
#include <hip/hip_runtime.h>
#include <float.h>
#include <stdint.h>

// ---------------------------------------------------------------------------
// KNN cosine-similarity classify on MI455X (gfx1250, wave32).
// Stage 1: async global->LDS staged GEMV via V_WMMA_F32_16X16X4_F32 (fp32).
//          Coalesced 512B/instr async copies (ASYNCcnt) + conflict-free
//          ds_load_b64 fragment reads; double-buffered per wave.
// Stage 2: 11-pass argmax top-k + 9-NN vote, single block.
// ---------------------------------------------------------------------------

typedef __attribute__((ext_vector_type(2))) float v2f;
typedef __attribute__((ext_vector_type(8))) float v8f;

#define DIMK 512
#define NCLS 100
#define TOPK 11                        // reference returns top (1+K) = 11 values
#define WAVES_PER_BLOCK 8
#define CHUNK_COLS 64                  // K-columns staged per chunk
#define NCHUNKS (DIMK / CHUNK_COLS)    // 8
#define ROW_STRIDE 68                  // floats: 272B row = 16B-aligned + bank-conflict-free
#define BUF_FLOATS (16 * ROW_STRIDE)   // one 16x64 tile buffer (padded)
#define WAVE_FLOATS (2 * BUF_FLOATS)   // double buffer
#define LDS_BYTES (WAVES_PER_BLOCK * WAVE_FLOATS * 4)   // 69632 B / block

// ---------------------------------------------------------------------------
// Stage 1: one wave computes cosine similarity for a 16-row tile.
//   A (16x4 f32): lane L -> row L%16; K = {0,1} (lanes 0..15) / {2,3} (16..31)
//   B (4x16 f32): query broadcast across all 16 N columns (same K mapping)
//   D (16x16 f32): lanes 0/16 hold rows 0..7 / 8..15 in acc[0..7]
// ---------------------------------------------------------------------------
__global__ __launch_bounds__(256) void cos_gemv_wmma(
    const float* __restrict__ coll, const float* __restrict__ q,
    float* __restrict__ cosv, int N)
{
    extern __shared__ float smemf[];

    const int wave = (int)(threadIdx.x >> 5);
    const int lane = (int)(threadIdx.x & 31);
    const int tile = (int)blockIdx.x * WAVES_PER_BLOCK + wave;
    const int numTiles = (N + 15) >> 4;
    if (tile >= numTiles) return;      // wave-uniform: EXEC stays all-ones for WMMA

    const int base  = tile << 4;
    const int m     = lane & 15;       // row within tile (fragment role)
    const int half  = lane >> 4;       // 0: K={0,1}, 1: K={2,3}
    const int kk    = half << 1;
    const int rlane = lane & 15;       // column-group within a row for async copy
    const int waveBase = wave * WAVE_FLOATS;

    // Issue one 16x64 fp32 chunk as 8 async B128 copies (512B each, coalesced):
    // instruction i moves rows 2i (lanes 0..15) and 2i+1 (lanes 16..31),
    // each lane carrying 16 contiguous bytes of its row.
    auto issue_chunk = [&](int c, int b) {
        const int bb = waveBase + b * BUF_FLOATS;
#pragma unroll
        for (int i = 0; i < 8; ++i) {
            const int rr = 2 * i + half;
            int grow = base + rr;
            if (grow >= N) grow = N - 1;            // stay in bounds (result unused)
            const float* gp = coll + (size_t)grow * DIMK + c * CHUNK_COLS + rlane * 4;
            const unsigned lo =
                (unsigned)(uintptr_t)(&smemf[bb + rr * ROW_STRIDE + rlane * 4]);
            const unsigned long long ga = (unsigned long long)(uintptr_t)gp;
            asm volatile("global_load_async_to_lds_b128 %0, %1, off"
                         :: "v"(lo), "v"(ga) : "memory");
        }
    };

    v8f acc = {};
    float rpart = 0.0f;   // per-lane partial sum of row x^2 (this lane's K residues)
    float qpart = 0.0f;   // per-lane partial sum of q^2

    issue_chunk(0, 0);
    for (int c = 0; c < NCHUNKS; ++c) {
        const int b = c & 1;
        if (c + 1 < NCHUNKS) {
            // buffer (c+1)&1 was last read in the previous iteration; fence DS reads
            // before the async engine overwrites it (separate pipes/counters).
            asm volatile("s_wait_dscnt 0x0" ::: "memory");
            issue_chunk(c + 1, (c + 1) & 1);
            asm volatile("s_wait_asynccnt 0x8" ::: "memory"); // chunk c landed
        } else {
            asm volatile("s_wait_asynccnt 0x0" ::: "memory");
        }

        const int bb = waveBase + b * BUF_FLOATS;
        const float* qc = q + c * CHUNK_COLS;
#pragma unroll
        for (int p = 0; p < CHUNK_COLS / 4; ++p) {
            const int k = (p << 2) + kk;
            // A fragment from LDS: ds_load_b64, conflict-free (stride 68 floats)
            v2f av = *(const v2f*)&smemf[bb + m * ROW_STRIDE + k];
            // B fragment: query broadcast (2KB, cache-hot)
            v2f bv = *(const v2f*)(qc + k);
            acc = __builtin_amdgcn_wmma_f32_16x16x4_f32(
                false, av, false, bv, (short)0, acc, false, false);
            rpart += av.x * av.x + av.y * av.y;
            qpart += bv.x * bv.x + bv.y * bv.y;
        }
    }

    // combine K-residue halves (wave32): lane m <-> lane m+16 hold the two halves
    const float rn_all = rpart + __shfl_xor(rpart, 16, 32);  // rownorm2[lane%16]
    const float qn2    = qpart + __shfl_xor(qpart, 16, 32);  // full ||q||^2
    const float invq   = rsqrtf(qn2 + 1e-12f);

    // lanes 0 and 16 hold rows base+0..7 / base+8..15 in acc[0..7] (any column N)
#pragma unroll
    for (int r = 0; r < 8; ++r) {
        const float rn   = __shfl(rn_all, r + (half << 3), 32);
        const float val  = acc[r] * rsqrtf(rn + 1e-12f) * invq;
        const int   orow = base + r + (half << 3);
        if (m == 0 && orow < N) cosv[orow] = val;
    }
}

// ---------------------------------------------------------------------------
// Stage 2: single block. 11 argmax passes (tie -> lower index, matches
// lax.top_k ordering), then majority vote over labels[idx[1..9]].
// cosv is fully rewritten by stage 1 every launch, so destructive selection
// is graph-replay safe.
// ---------------------------------------------------------------------------
__global__ __launch_bounds__(1024) void topk_vote(
    float* __restrict__ cosv, const int* __restrict__ labels,
    float* __restrict__ out, int N)
{
    __shared__ float sval[1024];
    __shared__ int   sidx[1024];
    __shared__ float topv[TOPK];
    __shared__ int   topi[TOPK];

    const int tid = (int)threadIdx.x;

    for (int t = 0; t < TOPK; ++t) {
        float bv = -FLT_MAX;
        int   bi = 0x7fffffff;
        for (int i = tid; i < N; i += 1024) {
            const float v = cosv[i];
            if (v > bv || (v == bv && i < bi)) { bv = v; bi = i; }
        }
        sval[tid] = bv; sidx[tid] = bi;
        __syncthreads();
        for (int s = 512; s > 0; s >>= 1) {
            if (tid < s) {
                const float ov = sval[tid + s];
                const int   oi = sidx[tid + s];
                if (ov > sval[tid] || (ov == sval[tid] && oi < sidx[tid])) {
                    sval[tid] = ov; sidx[tid] = oi;
                }
            }
            __syncthreads();
        }
        if (tid == 0) {
            topv[t] = sval[0];
            topi[t] = sidx[0];
            cosv[sidx[0]] = -FLT_MAX;   // knock out winner for next pass
        }
        __syncthreads();
    }

    if (tid == 0) {
        // reference: neigh_idx = idx[1:K] with K=10 -> 9 neighbors
        int preds[9];
        for (int j = 0; j < 9; ++j) preds[j] = labels[topi[1 + j]];
        int counts[NCLS];
        for (int c = 0; c < NCLS; ++c) counts[c] = 0;
        for (int j = 0; j < 9; ++j) counts[preds[j]]++;
        int bestc = 0;
        for (int c = 1; c < NCLS; ++c)
            if (counts[c] > counts[bestc]) bestc = c;   // first max, like jnp.argmax
        int cp = 0;
        for (int j = 8; j >= 0; --j)
            if (preds[j] == bestc) cp = j;              // first match position

        for (int t = 0; t < TOPK; ++t) out[t] = topv[t];
        out[TOPK]     = (float)bestc;                   // pred_int (as float output)
        out[TOPK + 1] = topv[1 + cp];                   // confidence
    }
}

// ---------------------------------------------------------------------------
extern "C" void kernel_launch(void* const* d_in, const int* in_sizes, int n_in,
                              void* d_out, int out_size, void* d_ws, size_t ws_size,
                              hipStream_t stream)
{
    const float* q      = (const float*)d_in[0];   // embedding (512,) f32
    const float* coll   = (const float*)d_in[1];   // collection (N,512) f32
    const int*   labels = (const int*)d_in[2];     // labels (N,) int
    const int N = in_sizes[2];

    float* cosv = (float*)d_ws;                    // N floats of scratch

    const int numTiles = (N + 15) / 16;
    const int blocks   = (numTiles + WAVES_PER_BLOCK - 1) / WAVES_PER_BLOCK;
    cos_gemv_wmma<<<blocks, 256, LDS_BYTES, stream>>>(coll, q, cosv, N);
    topk_vote<<<1, 1024, 0, stream>>>(cosv, labels, (float*)d_out, N);
}